// Qu_conv_mutual_17454747091658
// MI455X (gfx1250) — compile-verified
//
#include <hip/hip_runtime.h>
#include <math.h>

typedef float v2f __attribute__((ext_vector_type(2)));
typedef float v8f __attribute__((ext_vector_type(8)));

#define HP 16
#define WMUL 0.6324555320336759f   // sqrt(2)/sqrt(5)
#define ND_WAVES 32
#define NT_WAVES 128

// ---------------------------------------------------------------------------
// Kernel 1: Gram matrix G = E^T E over gathered embedding rows, via
// V_WMMA_F32_16X16X4_F32.  Each wave processes chunks of 4 tokens:
//   A (16x4) = chunk^T,  B (4x16) = chunk.  Per the CDNA5 ISA VGPR layouts
// the A and B register images coincide for E^T E, so one v2f per lane feeds
// both operands.  C layout: vgpr v, lane L -> row v + 8*(L>=16), col L%16.
// Loads are unconditional (row 0 is valid memory); padding_idx==0 zeroing is
// a VALU select so EXEC stays all-ones throughout (WMMA requirement).
// ---------------------------------------------------------------------------
__global__ __launch_bounds__(256) void gram_wmma_kernel(
    const float* __restrict__ table, const int* __restrict__ idx,
    int n_tokens, float* __restrict__ part, int n_waves)
{
  int tid  = blockIdx.x * blockDim.x + threadIdx.x;
  int wave = tid >> 5;
  int lane = tid & 31;
  int hi   = lane >> 4;     // 0: holds K=0,1   1: holds K=2,3
  int col  = lane & 15;
  int nchunk = n_tokens >> 2;

  v8f acc = {};
  for (int c = wave; c < nchunk; c += n_waves) {
    int base = 4 * c + 2 * hi;
    int t0 = idx[base];
    int t1 = idx[base + 1];
    float x0 = table[t0 * HP + col];   // unconditional: row 0 exists
    float x1 = table[t1 * HP + col];
    v2f a;
    a.x = (t0 == 0) ? 0.0f : x0;       // padding_idx = 0 -> v_cndmask
    a.y = (t1 == 0) ? 0.0f : x1;
    acc = __builtin_amdgcn_wmma_f32_16x16x4_f32(
        false, a, false, a, (short)0, acc, false, false);
  }
  float* dst = part + wave * 256;
#pragma unroll
  for (int v = 0; v < 8; ++v)
    dst[(v + 8 * hi) * 16 + col] = acc[v];
}

// ---------------------------------------------------------------------------
// Kernel 2: everything else in one 256-thread workgroup.
// ---------------------------------------------------------------------------
struct c2 { float x, y; };
__device__ inline c2 cmul(c2 a, c2 b) { return { a.x*b.x - a.y*b.y, a.x*b.y + a.y*b.x }; }
__device__ inline c2 cadd(c2 a, c2 b) { return { a.x + b.x, a.y + b.y }; }

__global__ __launch_bounds__(256) void tail_kernel(
    const float* __restrict__ gdP, const float* __restrict__ gtP,
    const float* __restrict__ dc1_w, const float* __restrict__ dc1_b,
    const float* __restrict__ tc1_w, const float* __restrict__ tc1_b,
    const float* __restrict__ dc2_w, const float* __restrict__ dc2_b,
    const float* __restrict__ tc2_w, const float* __restrict__ tc2_b,
    const float* __restrict__ qw,
    const float* __restrict__ fc1_w, const float* __restrict__ fc1_b,
    const float* __restrict__ fc2_w, const float* __restrict__ fc2_b,
    float* __restrict__ out)
{
  __shared__ float  Gd[256], Gt[256];
  __shared__ float  cdm[4][16], ctm[4][16];
  __shared__ c2     gate_s[8][4];
  __shared__ float2 buf[16][256];        // 16 state vectors x 256 amplitudes
  __shared__ float  d2in[16], t2in[16];  // c c^T per branch (4x4)
  __shared__ float  diagAB[2][16];
  __shared__ float  dacc[256];
  __shared__ float  trs[2];

  const int m = threadIdx.x;

  // ---- reduce per-wave Gram partials -------------------------------------
  {
    float s = 0.f;
    for (int w = 0; w < ND_WAVES; ++w) s += gdP[w * 256 + m];
    Gd[m] = s;
    s = 0.f;
    for (int w = 0; w < NT_WAVES; ++w) s += gtP[w * 256 + m];
    Gt[m] = s;
  }
  __syncthreads();

  // ---- conv1d 'same' (k=4, pad left1/right2): c[o,p] ---------------------
  if (m < 128) {
    const float* G  = (m < 64) ? Gd : Gt;
    const float* w1 = (m < 64) ? dc1_w : tc1_w;
    const float* b1 = (m < 64) ? dc1_b : tc1_b;
    int mm = m & 63;
    int o = mm >> 4, p = mm & 15;
    float s = b1[o];
    for (int k = 0; k < 4; ++k) {
      int q = p + k - 1;
      if (q >= 0 && q < 16)
        for (int i = 0; i < 16; ++i)
          s += G[i * 16 + q] * w1[(o * 16 + i) * 4 + k];
    }
    if (m < 64) cdm[o][p] = s; else ctm[o][p] = s;
  }
  __syncthreads();

  // ---- c c^T (4x4) per branch -------------------------------------------
  if (m < 32) {
    int which = m >> 4;
    int mm = m & 15;
    int r = mm >> 2, c = mm & 3;
    float s = 0.f;
    for (int p = 0; p < 16; ++p)
      s += which ? ctm[r][p] * ctm[c][p] : cdm[r][p] * cdm[c][p];
    if (which) t2in[mm] = s; else d2in[mm] = s;
  }
  __syncthreads();

  // ---- traces + single-qubit gates G_q = Rz Ry Rx ------------------------
  if (m == 0) {
    trs[0] = d2in[0] + d2in[5] + d2in[10] + d2in[15];  // tr(c^T c) = tr(c c^T)
    trs[1] = t2in[0] + t2in[5] + t2in[10] + t2in[15];
    for (int q = 0; q < 8; ++q) {
      float tx = qw[q]      * WMUL * 0.5f;
      float ty = qw[8 + q]  * WMUL * 0.5f;
      float tz = qw[16 + q] * WMUL * 0.5f;
      float cx = cosf(tx), sx = sinf(tx);
      float cy = cosf(ty), sy = sinf(ty);
      float cz = cosf(tz), sz = sinf(tz);
      c2 Rx[4] = { {cx, 0.f}, {0.f, -sx}, {0.f, -sx}, {cx, 0.f} };
      c2 Ry[4] = { {cy, 0.f}, {-sy, 0.f}, {sy, 0.f},  {cy, 0.f} };
      c2 z0 = { cz, -sz }, z1 = { cz, sz };       // Rz diagonal
      c2 M[4];
      M[0] = cadd(cmul(Ry[0], Rx[0]), cmul(Ry[1], Rx[2]));
      M[1] = cadd(cmul(Ry[0], Rx[1]), cmul(Ry[1], Rx[3]));
      M[2] = cadd(cmul(Ry[2], Rx[0]), cmul(Ry[3], Rx[2]));
      M[3] = cadd(cmul(Ry[2], Rx[1]), cmul(Ry[3], Rx[3]));
      gate_s[q][0] = cmul(z0, M[0]);
      gate_s[q][1] = cmul(z0, M[1]);
      gate_s[q][2] = cmul(z1, M[2]);
      gate_s[q][3] = cmul(z1, M[3]);
    }
  }
  __syncthreads();

  // ---- CNOT-ring permutation sigma(m) (E scatter target) -----------------
  int sig = m;
  for (int q = 0; q < 7; ++q)
    if ((sig >> (7 - q)) & 1) sig ^= (1 << (6 - q));
  if (sig & 1) sig ^= 128;                         // CNOT(wire7 -> wire0)

  const int p_hi = m >> 4, p_lo = m & 15;

  // ---- two passes: rho_d (x) rho_t   and   rho_t (x) rho_d ---------------
  for (int pass = 0; pass < 2; ++pass) {
    float vr[16], vi[16];
#pragma unroll
    for (int v = 0; v < 16; ++v) {
      int a = v >> 2, b = v & 3;
      float val = (pass == 0) ? cdm[a][p_hi] * ctm[b][p_lo]
                              : ctm[a][p_hi] * cdm[b][p_lo];
      vr[v] = val; vi[v] = 0.f;
    }
    // apply L, permute by E, apply L again
    for (int stage = 0; stage < 2; ++stage) {
      for (int q = 0; q < 8; ++q) {
        __syncthreads();
#pragma unroll
        for (int v = 0; v < 16; ++v) buf[v][m] = make_float2(vr[v], vi[v]);
        __syncthreads();
        int b = 7 - q;
        int mybit = (m >> b) & 1;
        int par = m ^ (1 << b);
        c2 go = gate_s[q][mybit * 2 + mybit];
        c2 gp = gate_s[q][mybit * 2 + (mybit ^ 1)];
#pragma unroll
        for (int v = 0; v < 16; ++v) {
          float2 pv = buf[v][par];
          float orr = vr[v], ori = vi[v];
          vr[v] = go.x * orr - go.y * ori + gp.x * pv.x - gp.y * pv.y;
          vi[v] = go.x * ori + go.y * orr + gp.x * pv.y + gp.y * pv.x;
        }
      }
      if (stage == 0) {
        __syncthreads();
#pragma unroll
        for (int v = 0; v < 16; ++v) buf[v][sig] = make_float2(vr[v], vi[v]);
        __syncthreads();
#pragma unroll
        for (int v = 0; v < 16; ++v) { float2 t = buf[v][m]; vr[v] = t.x; vi[v] = t.y; }
      }
    }
    // |amplitude|^2 summed over the 16 rank-1 components
    float s = 0.f;
#pragma unroll
    for (int v = 0; v < 16; ++v) s += vr[v] * vr[v] + vi[v] * vi[v];
    dacc[m] = s;
    __syncthreads();
    if (m < 16) {            // partial trace over second subsystem: diag only
      float d = 0.f;
      for (int j = 0; j < 16; ++j) d += dacc[m * 16 + j];
      diagAB[pass][m] = d;
    }
    __syncthreads();
  }

  // ---- measurement + conv2d + FC tail (tiny; serial on thread 0) ---------
  if (m == 0) {
    float inv = 1.0f / (trs[0] * trs[1]);
    float dmv[4], tmv[4];
    for (int i = 0; i < 4; ++i) {
      float sd = 0.f, st = 0.f;
      for (int a = 0; a < 16; ++a) {
        float sign = ((a >> (3 - i)) & 1) ? -1.f : 1.f;
        sd += sign * diagAB[0][a];
        st += sign * diagAB[1][a];
      }
      dmv[i] = sd * inv;
      tmv[i] = st * inv;
    }
    float flat[128];
    for (int br = 0; br < 2; ++br) {
      const float* w2  = br ? tc2_w : dc2_w;
      const float* b2  = br ? tc2_b : dc2_b;
      const float* xin = br ? t2in  : d2in;   // channel 0
      const float* mv  = br ? tmv   : dmv;    // channel 1 = outer(mv,mv)
      for (int o = 0; o < 4; ++o)
        for (int r = 0; r < 4; ++r)
          for (int s = 0; s < 4; ++s) {
            float acc = b2[o];
            for (int u = 0; u < 3; ++u)
              for (int vv = 0; vv < 3; ++vv) {
                int rr = r + u - 1, ss = s + vv - 1;
                if (rr < 0 || rr > 3 || ss < 0 || ss > 3) continue;
                acc += xin[rr * 4 + ss] * w2[((o * 2 + 0) * 3 + u) * 3 + vv]
                     + mv[rr] * mv[ss]  * w2[((o * 2 + 1) * 3 + u) * 3 + vv];
              }
            flat[br * 64 + o * 16 + r * 4 + s] = acc;
          }
    }
    float h[32];
    for (int j = 0; j < 32; ++j) {
      float s = fc1_b[j];
      for (int i = 0; i < 128; ++i) s += flat[i] * fc1_w[j * 128 + i];
      h[j] = s > 0.f ? s : 0.01f * s;
    }
    float o2 = fc2_b[0];
    for (int j = 0; j < 32; ++j) o2 += h[j] * fc2_w[j];
    out[0] = o2 > 0.f ? o2 : 0.01f * o2;
  }
}

// ---------------------------------------------------------------------------
extern "C" void kernel_launch(void* const* d_in, const int* in_sizes, int n_in,
                              void* d_out, int out_size, void* d_ws, size_t ws_size,
                              hipStream_t stream) {
  const float* embed_drug   = (const float*)d_in[0];
  const float* embed_target = (const float*)d_in[1];
  const float* dc1_w = (const float*)d_in[2];
  const float* dc1_b = (const float*)d_in[3];
  const float* tc1_w = (const float*)d_in[4];
  const float* tc1_b = (const float*)d_in[5];
  const float* dc2_w = (const float*)d_in[6];
  const float* dc2_b = (const float*)d_in[7];
  const float* tc2_w = (const float*)d_in[8];
  const float* tc2_b = (const float*)d_in[9];
  const float* qw    = (const float*)d_in[10];
  const float* fc1_w = (const float*)d_in[11];
  const float* fc1_b = (const float*)d_in[12];
  const float* fc2_w = (const float*)d_in[13];
  const float* fc2_b = (const float*)d_in[14];
  const int*   drug   = (const int*)d_in[15];
  const int*   target = (const int*)d_in[16];

  int nd = in_sizes[15];   // 4096
  int nt = in_sizes[16];   // 32768

  float* ws  = (float*)d_ws;
  float* gdP = ws;                     // ND_WAVES * 256 floats
  float* gtP = ws + ND_WAVES * 256;    // NT_WAVES * 256 floats

  gram_wmma_kernel<<<ND_WAVES / 8, 256, 0, stream>>>(embed_drug, drug, nd, gdP, ND_WAVES);
  gram_wmma_kernel<<<NT_WAVES / 8, 256, 0, stream>>>(embed_target, target, nt, gtP, NT_WAVES);
  tail_kernel<<<1, 256, 0, stream>>>(gdP, gtP,
                                     dc1_w, dc1_b, tc1_w, tc1_b,
                                     dc2_w, dc2_b, tc2_w, tc2_b,
                                     qw, fc1_w, fc1_b, fc2_w, fc2_b,
                                     (float*)d_out);
}